// HomogeneousGraphSAGE_43130061586801
// MI455X (gfx1250) — compile-verified
//
#include <hip/hip_runtime.h>
#include <hip/hip_bf16.h>

#define N_NODES 100000
#define FEAT 64
#define NCLS 40
#define BN_EPS 1e-5f

typedef float v2f __attribute__((ext_vector_type(2)));
typedef float v8f __attribute__((ext_vector_type(8)));

// ---------------------------------------------------------------- utilities
__global__ void zero_int_kernel(int* __restrict__ p, int n) {
    int i = blockIdx.x * blockDim.x + threadIdx.x;
    if (i < n) p[i] = 0;
}

__global__ void zero_f_kernel(float* __restrict__ p, int n) {
    int i = blockIdx.x * blockDim.x + threadIdx.x;
    if (i < n) p[i] = 0.0f;
}

// ----------------------------------------------------------- CSR build (once)
__global__ __launch_bounds__(256)
void count_kernel(const long long* __restrict__ dst, int* __restrict__ degC, int E) {
    int e = blockIdx.x * blockDim.x + threadIdx.x;
    if (e < E) atomicAdd(&degC[(int)dst[e]], 1);
}

// block-local exclusive scan of 1024 elements per block (256 thr x 4 elems)
__global__ __launch_bounds__(256)
void scan_block_kernel(const int* __restrict__ degC, int* __restrict__ offs,
                       int* __restrict__ blockSums, int n) {
    __shared__ int sT[256];
    const int t = threadIdx.x;
    const int base = blockIdx.x * 1024 + t * 4;
    int d0 = (base + 0 < n) ? degC[base + 0] : 0;
    int d1 = (base + 1 < n) ? degC[base + 1] : 0;
    int d2 = (base + 2 < n) ? degC[base + 2] : 0;
    int d3 = (base + 3 < n) ? degC[base + 3] : 0;
    int tot = d0 + d1 + d2 + d3;
    sT[t] = tot;
    __syncthreads();
    for (int off = 1; off < 256; off <<= 1) {
        int v = (t >= off) ? sT[t - off] : 0;
        __syncthreads();
        sT[t] += v;
        __syncthreads();
    }
    int excl = sT[t] - tot;
    if (base + 0 < n) offs[base + 0] = excl;
    if (base + 1 < n) offs[base + 1] = excl + d0;
    if (base + 2 < n) offs[base + 2] = excl + d0 + d1;
    if (base + 3 < n) offs[base + 3] = excl + d0 + d1 + d2;
    if (t == 255) blockSums[blockIdx.x] = sT[255];
}

__global__ void scan_top_kernel(int* __restrict__ blockSums, int nb) {
    if (threadIdx.x == 0) {
        int acc = 0;
        for (int i = 0; i < nb; ++i) {
            int v = blockSums[i];
            blockSums[i] = acc;
            acc += v;
        }
    }
}

__global__ __launch_bounds__(256)
void scan_add_kernel(int* __restrict__ offs, const int* __restrict__ blockSums,
                     int* __restrict__ cursor, int n) {
    int i = blockIdx.x * blockDim.x + threadIdx.x;
    if (i < n) {
        int v = offs[i] + blockSums[i >> 10];
        offs[i] = v;
        cursor[i] = v;
    }
}

__global__ __launch_bounds__(256)
void fill_kernel(const long long* __restrict__ src, const long long* __restrict__ dst,
                 int* __restrict__ cursor, int* __restrict__ bucket, int E) {
    int e = blockIdx.x * blockDim.x + threadIdx.x;
    if (e < E) {
        int pos = atomicAdd(&cursor[(int)dst[e]], 1);
        bucket[pos] = (int)src[e];
    }
}

// ------------------------------------------------- atomic-free mean aggregate
// one wave per node: each lane owns 2 consecutive features (float2 loads).
__global__ __launch_bounds__(256)
void aggregate_kernel(const float* __restrict__ x, const int* __restrict__ offs,
                      const int* __restrict__ degC, const int* __restrict__ bucket,
                      float* __restrict__ agg) {
    const int wave = threadIdx.x >> 5;
    const int lane = threadIdx.x & 31;
    const int n = blockIdx.x * 8 + wave;   // N % 8 == 0
    const int start = offs[n];
    const int cnt = degC[n];
    float ax = 0.0f, ay = 0.0f;
    for (int i = 0; i < cnt; ++i) {
        int s = bucket[start + i];
        const float2 v = *(const float2*)(x + (size_t)s * FEAT + lane * 2);
        ax += v.x;
        ay += v.y;
    }
    float inv = 1.0f / fmaxf((float)cnt, 1.0f);
    float2 r;
    r.x = ax * inv;
    r.y = ay * inv;
    *(float2*)(agg + (size_t)n * FEAT + lane * 2) = r;
}

// ------------------------------------------------ fused SAGE linear (WMMA)
// out[n,:] = agg[n,:] @ Wl^T + bl + xin[n,:] @ Wr^T
// 32 nodes per block, 8 waves = 2 row-tiles x 4 col-tiles of 16x16.
__global__ __launch_bounds__(256)
void sage_gemm_kernel(const float* __restrict__ xin,   // [N,64]
                      const float* __restrict__ agg,   // [N,64] (pre-divided)
                      const float* __restrict__ Wl,    // [64,64] row-major
                      const float* __restrict__ bl,    // [64]
                      const float* __restrict__ Wr,    // [64,64]
                      float* __restrict__ out) {       // [N,64]
    __shared__ float sAgg[32 * 66];
    __shared__ float sX[32 * 66];

    const int node0 = blockIdx.x * 32;
    const int t = threadIdx.x;

    for (int i = t; i < 512; i += 256) {
        int node = i >> 4;
        int fb = (i & 15) * 4;
        const float4 a = *(const float4*)(agg + (size_t)(node0 + node) * FEAT + fb);
        const float4 b = *(const float4*)(xin + (size_t)(node0 + node) * FEAT + fb);
        float* pa = sAgg + node * 66 + fb;
        float* pb = sX + node * 66 + fb;
        pa[0] = a.x; pa[1] = a.y; pa[2] = a.z; pa[3] = a.w;
        pb[0] = b.x; pb[1] = b.y; pb[2] = b.z; pb[3] = b.w;
    }
    __syncthreads();

    const int wave = t >> 5;
    const int lane = t & 31;
    const int half = lane >> 4;
    const int l16 = lane & 15;
    const int mBase = (wave >> 2) * 16;   // 0 or 16
    const int jBase = (wave & 3) * 16;    // 0,16,32,48

    v8f c = {};
    const float* aggRow = sAgg + (mBase + l16) * 66;
    const float* xRow   = sX   + (mBase + l16) * 66;
    const float* wlRow  = Wl + (size_t)(jBase + l16) * FEAT;
    const float* wrRow  = Wr + (size_t)(jBase + l16) * FEAT;

#pragma unroll
    for (int kb = 0; kb < FEAT; kb += 4) {
        int k = kb + 2 * half;
        v2f aA = *(const v2f*)(aggRow + k);
        v2f bL = *(const v2f*)(wlRow + k);
        c = __builtin_amdgcn_wmma_f32_16x16x4_f32(false, aA, false, bL,
                                                  (short)0, c, false, false);
        v2f aX = *(const v2f*)(xRow + k);
        v2f bR = *(const v2f*)(wrRow + k);
        c = __builtin_amdgcn_wmma_f32_16x16x4_f32(false, aX, false, bR,
                                                  (short)0, c, false, false);
    }

    const float bias = bl[jBase + l16];
    const int col = jBase + l16;
#pragma unroll
    for (int r = 0; r < 8; ++r) {
        int row = node0 + mBase + r + 8 * half;
        out[(size_t)row * FEAT + col] = c[r] + bias;
    }
}

// ------------------------------------------------------------- BN statistics
__global__ __launch_bounds__(256)
void bn_stats_kernel(const float* __restrict__ hpre,
                     float* __restrict__ sums, float* __restrict__ sumsq, int n) {
    __shared__ float sS[4][64];
    __shared__ float sS2[4][64];
    const int c = threadIdx.x & 63;
    const int g = threadIdx.x >> 6;
    const int row0 = blockIdx.x * 128;
    float s = 0.0f, s2 = 0.0f;
    for (int r = g; r < 128; r += 4) {
        int row = row0 + r;
        if (row < n) {
            float v = hpre[(size_t)row * FEAT + c];
            s += v;
            s2 += v * v;
        }
    }
    sS[g][c] = s;
    sS2[g][c] = s2;
    __syncthreads();
    if (threadIdx.x < 64) {
        float ts = sS[0][c] + sS[1][c] + sS[2][c] + sS[3][c];
        float ts2 = sS2[0][c] + sS2[1][c] + sS2[2][c] + sS2[3][c];
        unsafeAtomicAdd(sums + c, ts);
        unsafeAtomicAdd(sumsq + c, ts2);
    }
}

__global__ void bn_finalize_kernel(const float* __restrict__ sums,
                                   const float* __restrict__ sumsq,
                                   const float* __restrict__ gamma,
                                   const float* __restrict__ beta,
                                   float* __restrict__ scale,
                                   float* __restrict__ shift) {
    int c = threadIdx.x;
    if (c < FEAT) {
        const float invN = 1.0f / (float)N_NODES;
        float mean = sums[c] * invN;
        float var = sumsq[c] * invN - mean * mean;
        float sc = gamma[c] * rsqrtf(var + BN_EPS);
        scale[c] = sc;
        shift[c] = beta[c] - mean * sc;
    }
}

__global__ __launch_bounds__(256)
void bn_apply_relu_kernel(const float* __restrict__ hpre,
                          const float* __restrict__ scale,
                          const float* __restrict__ shift,
                          float* __restrict__ out, int total) {
    int i = blockIdx.x * blockDim.x + threadIdx.x;
    if (i < total) {
        int c = i & 63;
        float v = hpre[i] * scale[c] + shift[c];
        out[i] = fmaxf(v, 0.0f);
    }
}

// --------------------------------------------------------- decoder (WMMA)
// logits[n,:40] = h[n,:] @ Wd^T + bd ; 32 nodes/block, 6 waves = 2x3 tiles.
__global__ __launch_bounds__(192)
void decoder_kernel(const float* __restrict__ h,    // [N,64]
                    const float* __restrict__ Wd,   // [40,64]
                    const float* __restrict__ bd,   // [40]
                    float* __restrict__ logits) {   // [N,40]
    __shared__ float sH[32 * 66];
    const int node0 = blockIdx.x * 32;
    const int t = threadIdx.x;

    for (int i = t; i < 512; i += 192) {
        int node = i >> 4;
        int fb = (i & 15) * 4;
        const float4 a = *(const float4*)(h + (size_t)(node0 + node) * FEAT + fb);
        float* p = sH + node * 66 + fb;
        p[0] = a.x; p[1] = a.y; p[2] = a.z; p[3] = a.w;
    }
    __syncthreads();

    const int wave = t >> 5;      // 0..5
    const int lane = t & 31;
    const int half = lane >> 4;
    const int l16 = lane & 15;
    const int mBase = (wave / 3) * 16;  // 0 or 16
    const int jBase = (wave % 3) * 16;  // 0,16,32
    const int j = jBase + l16;

    v8f c = {};
    const float* hRow = sH + (mBase + l16) * 66;

#pragma unroll
    for (int kb = 0; kb < FEAT; kb += 4) {
        int k = kb + 2 * half;
        v2f aH = *(const v2f*)(hRow + k);
        v2f bW = {0.0f, 0.0f};
        if (j < NCLS) bW = *(const v2f*)(Wd + (size_t)j * FEAT + k);
        c = __builtin_amdgcn_wmma_f32_16x16x4_f32(false, aH, false, bW,
                                                  (short)0, c, false, false);
    }

    if (j < NCLS) {
        float bias = bd[j];
#pragma unroll
        for (int r = 0; r < 8; ++r) {
            int row = node0 + mBase + r + 8 * half;
            logits[(size_t)row * NCLS + j] = c[r] + bias;
        }
    }
}

// ------------------------------------------------------------------ driver
extern "C" void kernel_launch(void* const* d_in, const int* in_sizes, int n_in,
                              void* d_out, int out_size, void* d_ws, size_t ws_size,
                              hipStream_t stream) {
    const float* x = (const float*)d_in[0];
    const long long* edge = (const long long*)d_in[1];
    const float* W1l = (const float*)d_in[2];
    const float* b1l = (const float*)d_in[3];
    const float* W1r = (const float*)d_in[4];
    const float* g1  = (const float*)d_in[5];
    const float* be1 = (const float*)d_in[6];
    const float* W2l = (const float*)d_in[7];
    const float* b2l = (const float*)d_in[8];
    const float* W2r = (const float*)d_in[9];
    const float* g2  = (const float*)d_in[10];
    const float* be2 = (const float*)d_in[11];
    const float* Wd  = (const float*)d_in[12];
    const float* bd  = (const float*)d_in[13];

    const int E = in_sizes[1] / 2;
    const long long* src = edge;
    const long long* dst = edge + E;

    // workspace layout: ints first, then floats (all 16B aligned by construction)
    int* ip = (int*)d_ws;
    int* degC      = ip;                    // N
    int* offs      = degC + N_NODES;        // N
    int* cursor    = offs + N_NODES;        // N
    int* blockSums = cursor + N_NODES;      // 128
    int* bucket    = blockSums + 128;       // E
    float* agg   = (float*)(bucket + E);    // N*64
    float* hpre  = agg + (size_t)N_NODES * FEAT;
    float* h1    = hpre + (size_t)N_NODES * FEAT;
    float* stats = h1 + (size_t)N_NODES * FEAT;   // sums|sumsq|scale|shift (256)

    float* logits = (float*)d_out;
    float* h2 = logits + (size_t)N_NODES * NCLS;  // final hidden -> output slot

    const int nGrid = (N_NODES + 255) / 256;          // 391
    const int eGrid = (E + 255) / 256;                // 6250
    const int scanBlocks = (N_NODES + 1023) / 1024;   // 98
    const int aggGrid = N_NODES / 8;                  // 12500, exact
    const int gemmGrid = N_NODES / 32;                // 3125, exact
    const int statsGrid = (N_NODES + 127) / 128;      // 782
    const int ewTotal = N_NODES * FEAT;
    const int ewGrid = (ewTotal + 255) / 256;

    // -------- CSR build (once; reused by both layers) --------
    zero_int_kernel<<<nGrid, 256, 0, stream>>>(degC, N_NODES);
    count_kernel<<<eGrid, 256, 0, stream>>>(dst, degC, E);
    scan_block_kernel<<<scanBlocks, 256, 0, stream>>>(degC, offs, blockSums, N_NODES);
    scan_top_kernel<<<1, 32, 0, stream>>>(blockSums, scanBlocks);
    scan_add_kernel<<<nGrid, 256, 0, stream>>>(offs, blockSums, cursor, N_NODES);
    fill_kernel<<<eGrid, 256, 0, stream>>>(src, dst, cursor, bucket, E);

    // ---------------- layer 1 ----------------
    zero_f_kernel<<<1, 256, 0, stream>>>(stats, 256);
    aggregate_kernel<<<aggGrid, 256, 0, stream>>>(x, offs, degC, bucket, agg);
    sage_gemm_kernel<<<gemmGrid, 256, 0, stream>>>(x, agg, W1l, b1l, W1r, hpre);
    bn_stats_kernel<<<statsGrid, 256, 0, stream>>>(hpre, stats, stats + 64, N_NODES);
    bn_finalize_kernel<<<1, 64, 0, stream>>>(stats, stats + 64, g1, be1,
                                             stats + 128, stats + 192);
    bn_apply_relu_kernel<<<ewGrid, 256, 0, stream>>>(hpre, stats + 128, stats + 192,
                                                     h1, ewTotal);

    // ---------------- layer 2 ----------------
    zero_f_kernel<<<1, 256, 0, stream>>>(stats, 256);
    aggregate_kernel<<<aggGrid, 256, 0, stream>>>(h1, offs, degC, bucket, agg);
    sage_gemm_kernel<<<gemmGrid, 256, 0, stream>>>(h1, agg, W2l, b2l, W2r, hpre);
    bn_stats_kernel<<<statsGrid, 256, 0, stream>>>(hpre, stats, stats + 64, N_NODES);
    bn_finalize_kernel<<<1, 64, 0, stream>>>(stats, stats + 64, g2, be2,
                                             stats + 128, stats + 192);
    bn_apply_relu_kernel<<<ewGrid, 256, 0, stream>>>(hpre, stats + 128, stats + 192,
                                                     h2, ewTotal);

    // ---------------- decoder ----------------
    decoder_kernel<<<gemmGrid, 192, 0, stream>>>(h2, Wd, bd, logits);
}